// MLPEncoder_85942295593121
// MI455X (gfx1250) — compile-verified
//
#include <hip/hip_runtime.h>
#include <hip/hip_bf16.h>
#include <math.h>

typedef __attribute__((ext_vector_type(2))) float v2f;
typedef __attribute__((ext_vector_type(8))) float v8f;

// ---------------------------------------------------------------------------
// Problem constants
// ---------------------------------------------------------------------------
#define BB   16
#define NN   128
#define FF   64
#define NH   64
#define HH   2
#define DD   64
#define EE   16256            // N*(N-1)
#define ROWS_BN   (BB * NN)   // 2048

// ---------------------------------------------------------------------------
// fp32 WMMA tile helpers (V_WMMA_F32_16X16X4_F32, wave32)
// A: 16xK row-major (lda), B: weight matrix W[out][in] row-major (ldb) used as
// B[k][n] = W[n][k]  =>  D[m][n] = sum_k A[m][k] * W[n][k]   (i.e. A @ W^T)
// Fragment layout per ISA 7.12.2: lane l holds k = 2*(l>=16)+v of row/col l&15
// ---------------------------------------------------------------------------
__device__ __forceinline__ v8f vzero8() {
  v8f z;
#pragma unroll
  for (int i = 0; i < 8; ++i) z[i] = 0.0f;
  return z;
}

template <int K>
__device__ __forceinline__ v8f wmma_tile(const float* A, int lda,
                                         const float* B, int ldb, v8f acc) {
  const int lane = threadIdx.x & 31;
  const int rc   = lane & 15;            // A row / B(=W) row (output col)
  const int koff = (lane >> 4) << 1;     // 0 or 2
  const float* ap = A + rc * lda + koff;
  const float* bp = B + rc * ldb + koff;
#pragma unroll
  for (int kb = 0; kb < K; kb += 4) {
    v2f a, b;
    a.x = ap[kb];     a.y = ap[kb + 1];
    b.x = bp[kb];     b.y = bp[kb + 1];
    acc = __builtin_amdgcn_wmma_f32_16x16x4_f32(false, a, false, b,
                                                (short)0, acc, false, false);
  }
  return acc;
}

// C/D layout: lane l -> column l&15, rows (l>=16 ? 8 : 0) + v, v = 0..7
// act: 0 = none, 1 = ELU, 2 = ReLU
__device__ __forceinline__ void store_tile(float* dst, int ldd, int n0, v8f acc,
                                           const float* bias, int act) {
  const int lane = threadIdx.x & 31;
  const int c  = lane & 15;
  const int r0 = (lane >> 4) * 8;
#pragma unroll
  for (int v = 0; v < 8; ++v) {
    float val = acc[v];
    if (bias) val += bias[n0 + c];
    if (act == 1) val = (val > 0.0f) ? val : expm1f(val);   // jax.nn.elu
    else if (act == 2) val = fmaxf(val, 0.0f);
    dst[(r0 + v) * ldd + n0 + c] = val;
  }
}

// ---------------------------------------------------------------------------
// Kernel 1: f_h MLP (elu,elu,linear) + proj.  64 rows/block, 16 rows/wave.
// ---------------------------------------------------------------------------
__global__ __launch_bounds__(128) void k_node_in(
    const float* __restrict__ x,
    const float* __restrict__ W1, const float* __restrict__ b1,
    const float* __restrict__ W2, const float* __restrict__ b2,
    const float* __restrict__ W3, const float* __restrict__ b3,
    const float* __restrict__ Wp,
    float* __restrict__ v_self, float* __restrict__ vproj) {
  __shared__ float lds[4][16 * 68];
  const int w    = threadIdx.x >> 5;
  const int row0 = blockIdx.x * 64 + w * 16;
  float* S = lds[w];
  v8f acc[4];

  // h1 = elu(x @ W1^T + b1)
#pragma unroll
  for (int nt = 0; nt < 4; ++nt)
    acc[nt] = wmma_tile<64>(x + row0 * FF, FF, W1 + nt * 16 * FF, FF, vzero8());
#pragma unroll
  for (int nt = 0; nt < 4; ++nt) store_tile(S, 68, nt * 16, acc[nt], b1, 1);
  __syncthreads();

  // h2 = elu(h1 @ W2^T + b2)
#pragma unroll
  for (int nt = 0; nt < 4; ++nt)
    acc[nt] = wmma_tile<64>(S, 68, W2 + nt * 16 * NH, NH, vzero8());
  __syncthreads();
#pragma unroll
  for (int nt = 0; nt < 4; ++nt) store_tile(S, 68, nt * 16, acc[nt], b2, 1);
  __syncthreads();

  // v_self = h2 @ W3^T + b3   (to global AND LDS for proj)
#pragma unroll
  for (int nt = 0; nt < 4; ++nt)
    acc[nt] = wmma_tile<64>(S, 68, W3 + nt * 16 * NH, NH, vzero8());
  __syncthreads();
#pragma unroll
  for (int nt = 0; nt < 4; ++nt) {
    store_tile(S, 68, nt * 16, acc[nt], b3, 0);
    store_tile(v_self + row0 * NH, NH, nt * 16, acc[nt], b3, 0);
  }
  __syncthreads();

  // v_proj = v_self @ proj^T   (128 output cols)
#pragma unroll
  for (int nt = 0; nt < 8; ++nt) {
    v8f a = wmma_tile<64>(S, 68, Wp + nt * 16 * NH, NH, vzero8());
    store_tile(vproj + row0 * 128, 128, nt * 16, a, nullptr, 0);
  }
}

// ---------------------------------------------------------------------------
// Kernel 2: edge attention + edge MLP + scatter.
// One block per (b, target j).  254 rows = 127 edges x 2 heads, 16-row tiles.
// fc1 A-fragments are fed straight from v_proj (global, L0/L2 resident) with
// the per-row alpha scale folded in -> no e_in staging pass, ~18 KB LDS/block.
// ---------------------------------------------------------------------------
__global__ __launch_bounds__(128) void k_edge(
    const float* __restrict__ vproj,
    const float* __restrict__ a_fwd, const float* __restrict__ a_bwd,
    const float* __restrict__ eW1, const float* __restrict__ eb1,
    const float* __restrict__ eW2, const float* __restrict__ eb2,
    float* __restrict__ alpha_out, float* __restrict__ vsoc_out) {
  __shared__ float sT[4][16 * 68];    // relu(fc1) tiles
  __shared__ float sAi[4][16];        // alpha_ij per row
  __shared__ float sAj[4][16];        // alpha_ji per row
  __shared__ float vacc[HH * DD];     // per-node accumulator
  const int tid  = threadIdx.x;
  const int w    = tid >> 5;
  const int lane = tid & 31;
  const int blk  = blockIdx.x;
  const int b    = blk >> 7;          // / 128
  const int j    = blk & 127;
  const float* vpb = vproj + (long)b * NN * 128;

  if (tid < HH * DD) vacc[tid] = 0.0f;
  __syncthreads();

  for (int tile = w; tile < 16; tile += 4) {   // exactly 4 iters per wave
    const int rowbase = tile * 16;

    // --- per-row attention scalars (lanes 0..15, one row each) -------------
    if (lane < 16) {
      const int grow   = rowbase + lane;
      const bool valid = grow < 254;
      float alpi = 0.0f, alpj = 0.0f;
      if (valid) {
        const int eidx = grow >> 1, h = grow & 1;
        const int i = eidx + (eidx >= j ? 1 : 0);
        const float* hs = vpb + i * 128 + h * DD;
        const float* ht = vpb + j * 128 + h * DD;
        const float* af = a_fwd + h * DD;
        const float* ab = a_bwd + h * DD;
        float ds = 0.0f, dt = 0.0f;
#pragma unroll 8
        for (int k = 0; k < DD; ++k) {
          ds = fmaf(hs[k], af[k], ds);
          dt = fmaf(ht[k], ab[k], dt);
        }
        float aij = (ds >= 0.0f ? ds : 0.2f * ds) * (1.0f / 500.0f);
        float aji = (dt >= 0.0f ? dt : 0.2f * dt) * (1.0f / 500.0f);
        float m   = fmaxf(aij, aji);
        float eij = expf(expf(aij - m));
        float eji = expf(expf(aji - m));
        float s   = eij + eji;
        alpi = eij / s;
        alpj = eji / s;
        const int pos = (j < i) ? j : j - 1;
        const int e   = i * (NN - 1) + pos;
        alpha_out[((long)b * EE + e) * HH + h] = alpi;
      }
      sAi[w][lane] = alpi;
      sAj[w][lane] = alpj;
    }
    __syncthreads();

    // --- per-lane A-row setup for fc1 (row r = lane&15, K split at 64) -----
    {
      const int r    = lane & 15;
      const int koff = (lane >> 4) << 1;
      const int grow = rowbase + r;
      const bool valid = grow < 254;
      const int eidx = grow >> 1, h = grow & 1;
      const int i  = valid ? (eidx + (eidx >= j ? 1 : 0)) : 0;
      const float ai = sAi[w][r];
      const float aj = sAj[w][r];
      const float* hsp = vpb + i * 128 + h * DD + koff;   // cols 0..63
      const float* htp = vpb + j * 128 + h * DD + koff;   // cols 64..127

      // --- fc1: relu([ai*h_src | aj*h_tgt] @ eW1^T + eb1), K = 128 ---------
#pragma unroll
      for (int nt = 0; nt < 4; ++nt) {
        const float* bp = eW1 + (nt * 16 + r) * 128 + koff;
        v8f acc = vzero8();
#pragma unroll
        for (int kb = 0; kb < 128; kb += 4) {
          v2f a, bv;
          if (kb < 64) { a.x = ai * hsp[kb];      a.y = ai * hsp[kb + 1]; }
          else         { a.x = aj * htp[kb - 64]; a.y = aj * htp[kb - 63]; }
          bv.x = bp[kb];  bv.y = bp[kb + 1];
          acc = __builtin_amdgcn_wmma_f32_16x16x4_f32(false, a, false, bv,
                                                      (short)0, acc, false, false);
        }
        store_tile(sT[w], 68, nt * 16, acc, eb1, 2);
      }
    }
    __syncthreads();

    // --- fc2 + alpha_ij scale + scatter into vacc --------------------------
    {
      const int c  = lane & 15;
      const int r0 = (lane >> 4) * 8;
#pragma unroll
      for (int nt = 0; nt < 4; ++nt) {
        v8f a2 = wmma_tile<64>(sT[w], 68, eW2 + nt * 16 * DD, DD, vzero8());
#pragma unroll
        for (int v = 0; v < 8; ++v) {
          const int r = r0 + v;
          const int h = (rowbase + r) & 1;
          const float val = (a2[v] + eb2[nt * 16 + c]) * sAi[w][r];
          atomicAdd(&vacc[h * DD + nt * 16 + c], val);   // ds_add_f32
        }
      }
    }
    __syncthreads();
  }

  if (tid < HH * DD) vsoc_out[(long)blk * (HH * DD) + tid] = vacc[tid];
}

// ---------------------------------------------------------------------------
// Kernel 3a: gv MLP (relu,relu,linear) on 4096 rows of 64.
// ---------------------------------------------------------------------------
__global__ __launch_bounds__(128) void k_gv(
    const float* __restrict__ vsoc,
    const float* __restrict__ W1, const float* __restrict__ b1,
    const float* __restrict__ W2, const float* __restrict__ b2,
    const float* __restrict__ W3, const float* __restrict__ b3,
    float* __restrict__ g) {
  __shared__ float lds[4][16 * 68];
  const int w    = threadIdx.x >> 5;
  const int row0 = blockIdx.x * 64 + w * 16;
  float* S = lds[w];
  v8f acc[4];

#pragma unroll
  for (int nt = 0; nt < 4; ++nt)
    acc[nt] = wmma_tile<64>(vsoc + row0 * DD, DD, W1 + nt * 16 * DD, DD, vzero8());
#pragma unroll
  for (int nt = 0; nt < 4; ++nt) store_tile(S, 68, nt * 16, acc[nt], b1, 2);
  __syncthreads();

#pragma unroll
  for (int nt = 0; nt < 4; ++nt)
    acc[nt] = wmma_tile<64>(S, 68, W2 + nt * 16 * NH, NH, vzero8());
  __syncthreads();
#pragma unroll
  for (int nt = 0; nt < 4; ++nt) store_tile(S, 68, nt * 16, acc[nt], b2, 2);
  __syncthreads();

#pragma unroll
  for (int nt = 0; nt < 4; ++nt) {
    v8f a = wmma_tile<64>(S, 68, W3 + nt * 16 * NH, NH, vzero8());
    store_tile(g + row0 * DD, DD, nt * 16, a, b3, 0);
  }
}

// ---------------------------------------------------------------------------
// Kernel 3b: cg MLP (elu,elu,linear), first layer K = 128 -> x_out.
// ---------------------------------------------------------------------------
__global__ __launch_bounds__(128) void k_node_out(
    const float* __restrict__ g,
    const float* __restrict__ W1, const float* __restrict__ b1,
    const float* __restrict__ W2, const float* __restrict__ b2,
    const float* __restrict__ W3, const float* __restrict__ b3,
    float* __restrict__ xout) {
  __shared__ float lds[4][16 * 68];
  const int w    = threadIdx.x >> 5;
  const int row0 = blockIdx.x * 64 + w * 16;
  float* S = lds[w];
  v8f acc[4];

#pragma unroll
  for (int nt = 0; nt < 4; ++nt)
    acc[nt] = wmma_tile<128>(g + row0 * 128, 128, W1 + nt * 16 * 128, 128, vzero8());
#pragma unroll
  for (int nt = 0; nt < 4; ++nt) store_tile(S, 68, nt * 16, acc[nt], b1, 1);
  __syncthreads();

#pragma unroll
  for (int nt = 0; nt < 4; ++nt)
    acc[nt] = wmma_tile<64>(S, 68, W2 + nt * 16 * NH, NH, vzero8());
  __syncthreads();
#pragma unroll
  for (int nt = 0; nt < 4; ++nt) store_tile(S, 68, nt * 16, acc[nt], b2, 1);
  __syncthreads();

#pragma unroll
  for (int nt = 0; nt < 4; ++nt) {
    v8f a = wmma_tile<64>(S, 68, W3 + nt * 16 * NH, NH, vzero8());
    store_tile(xout + row0 * NH, NH, nt * 16, a, b3, 0);
  }
}

// ---------------------------------------------------------------------------
// Launch.  Inputs: 0=inputs, 1=rel_rec (unused), 2=rel_send (unused),
// 3.. = params in dict order.  Outputs: x_out | v_self | alpha_ij.
// ---------------------------------------------------------------------------
extern "C" void kernel_launch(void* const* d_in, const int* in_sizes, int n_in,
                              void* d_out, int out_size, void* d_ws, size_t ws_size,
                              hipStream_t stream) {
  const float* x     = (const float*)d_in[0];
  const float* fhW1  = (const float*)d_in[3];
  const float* fhb1  = (const float*)d_in[4];
  const float* fhW2  = (const float*)d_in[5];
  const float* fhb2  = (const float*)d_in[6];
  const float* fhW3  = (const float*)d_in[7];
  const float* fhb3  = (const float*)d_in[8];
  const float* projW = (const float*)d_in[9];
  const float* a_fwd = (const float*)d_in[10];
  const float* a_bwd = (const float*)d_in[11];
  const float* eW1   = (const float*)d_in[12];
  const float* eb1   = (const float*)d_in[13];
  const float* eW2   = (const float*)d_in[14];
  const float* eb2   = (const float*)d_in[15];
  const float* gvW1  = (const float*)d_in[16];
  const float* gvb1  = (const float*)d_in[17];
  const float* gvW2  = (const float*)d_in[18];
  const float* gvb2  = (const float*)d_in[19];
  const float* gvW3  = (const float*)d_in[20];
  const float* gvb3  = (const float*)d_in[21];
  const float* cgW1  = (const float*)d_in[22];
  const float* cgb1  = (const float*)d_in[23];
  const float* cgW2  = (const float*)d_in[24];
  const float* cgb2  = (const float*)d_in[25];
  const float* cgW3  = (const float*)d_in[26];
  const float* cgb3  = (const float*)d_in[27];

  float* ws    = (float*)d_ws;
  float* vproj = ws;                       // 2048 * 128 floats
  float* vsoc  = ws + ROWS_BN * 128;       // 2048 * 128 floats
  float* g     = ws;                       // reuse vproj space (dead after k_edge)

  float* xout  = (float*)d_out;            // [2048, 64]
  float* vself = xout + ROWS_BN * NH;      // [2048, 64]
  float* alpha = xout + 2 * ROWS_BN * NH;  // [16, 16256, 2]

  k_node_in<<<ROWS_BN / 64, 128, 0, stream>>>(
      x, fhW1, fhb1, fhW2, fhb2, fhW3, fhb3, projW, vself, vproj);
  k_edge<<<BB * NN, 128, 0, stream>>>(
      vproj, a_fwd, a_bwd, eW1, eb1, eW2, eb2, alpha, vsoc);
  k_gv<<<(ROWS_BN * HH) / 64, 128, 0, stream>>>(
      vsoc, gvW1, gvb1, gvW2, gvb2, gvW3, gvb3, g);
  k_node_out<<<ROWS_BN / 64, 128, 0, stream>>>(
      g, cgW1, cgb1, cgW2, cgb2, cgW3, cgb3, xout);
}